// BaseModule_81956565942278
// MI455X (gfx1250) — compile-verified
//
#include <hip/hip_runtime.h>
#include <math.h>

#define Bb 2
#define Vv 3
#define Cc 64
#define Hh 256
#define Ww 256
#define Nn (Hh*Ww)
#define EPSV 1e-06f

typedef float v2f __attribute__((ext_vector_type(2)));
typedef float v8f __attribute__((ext_vector_type(8)));

// ---------------------------------------------------------------------------
// Kernel Z: zero d_out (feature+depth accumulators) and the weight grid in ws
// ---------------------------------------------------------------------------
__global__ void zero_kernel(float* out, int out_n, float* wsum, int ws_n) {
    int i = blockIdx.x * blockDim.x + threadIdx.x;
    if (i < out_n) out[i] = 0.0f;
    if (i < ws_n)  wsum[i] = 0.0f;
}

// ---------------------------------------------------------------------------
// Kernel M: per (b,v) fused transform  A3 = K * (dstRT*srcRTinv)[:3,:3] * K^-1,
//           t = K * (dstRT*srcRTinv)[:3,3].  Writes 12 floats per bv.
// ---------------------------------------------------------------------------
__global__ void mats_kernel(const float* __restrict__ Km,
                            const float* __restrict__ srcinv,
                            const float* __restrict__ dst,
                            float* __restrict__ mats) {
    int bv = threadIdx.x;
    if (bv >= Bb * Vv) return;
    int b = bv / Vv;
    const float* K = Km + b * 9;
    float k00=K[0],k01=K[1],k02=K[2],k10=K[3],k11=K[4],k12=K[5],k20=K[6],k21=K[7],k22=K[8];
    float det = k00*(k11*k22-k12*k21) - k01*(k10*k22-k12*k20) + k02*(k10*k21-k11*k20);
    float id = 1.0f / det;
    float Ki[3][3] = {
        {(k11*k22-k12*k21)*id, (k02*k21-k01*k22)*id, (k01*k12-k02*k11)*id},
        {(k12*k20-k10*k22)*id, (k00*k22-k02*k20)*id, (k02*k10-k00*k12)*id},
        {(k10*k21-k11*k20)*id, (k01*k20-k00*k21)*id, (k00*k11-k01*k10)*id}
    };
    float Kr[3][3] = {{k00,k01,k02},{k10,k11,k12},{k20,k21,k22}};
    const float* S = srcinv + bv * 16;   // (b,v,4,4)
    const float* D = dst + b * 16;       // (b,1,4,4)
    float M[3][4];
    for (int r = 0; r < 3; ++r)
        for (int c = 0; c < 4; ++c)
            M[r][c] = D[r*4+0]*S[0*4+c] + D[r*4+1]*S[1*4+c]
                    + D[r*4+2]*S[2*4+c] + D[r*4+3]*S[3*4+c];
    float KM[3][3], Kt[3];
    for (int r = 0; r < 3; ++r) {
        for (int c = 0; c < 3; ++c)
            KM[r][c] = Kr[r][0]*M[0][c] + Kr[r][1]*M[1][c] + Kr[r][2]*M[2][c];
        Kt[r] = Kr[r][0]*M[0][3] + Kr[r][1]*M[1][3] + Kr[r][2]*M[2][3];
    }
    float* o = mats + bv * 12;
    for (int r = 0; r < 3; ++r)
        for (int c = 0; c < 3; ++c)
            o[r*3+c] = KM[r][0]*Ki[0][c] + KM[r][1]*Ki[1][c] + KM[r][2]*Ki[2][c];
    o[9] = Kt[0]; o[10] = Kt[1]; o[11] = Kt[2];
}

// ---------------------------------------------------------------------------
// Kernel S: WMMA f32 16x16x4 projective transform + bilinear atomic splat.
// One thread per source pixel; each wave does 32 pixels via two WMMAs:
//   A-matrix rows 0..2 and 8..10 carry the 3x3 transform (duplicated) so that
//   lanes 0..15 read group-0 results (M=0..2) and lanes 16..31 read group-1
//   results (M=8..10) from their own D registers.
// ---------------------------------------------------------------------------
__global__ void splat_kernel(const float* __restrict__ feats,
                             const float* __restrict__ depths,
                             const float* __restrict__ mats,
                             float* __restrict__ out_f,   // B*V*C*N
                             float* __restrict__ out_d,   // B*V*N
                             float* __restrict__ wsum) {  // B*V*N
    const unsigned p    = blockIdx.x * blockDim.x + threadIdx.x;  // pixel id (exact cover)
    const unsigned lane = threadIdx.x & 31u;
    const unsigned bv   = p / Nn;          // wave-uniform (N % 32 == 0)
    const unsigned n    = p % Nn;
    const unsigned nbase = (p & ~31u) % Nn;

    const float* Mt = mats + bv * 12;
    const float T0 = Mt[9], T1 = Mt[10], T2 = Mt[11];

    // --- build WMMA A operand (16x4 f32: VGPR0 = K0|K2, VGPR1 = K1|K3) ---
    unsigned m = lane & 15u;
    int mr = (m < 3u) ? (int)m : ((m >= 8u && m < 11u) ? (int)(m - 8u) : -1);
    float a0 = 0.0f, a1 = 0.0f;
    if (mr >= 0) {
        a0 = (lane < 16u) ? Mt[mr*3 + 0] : Mt[mr*3 + 2];  // K=0 col | K=2 col
        a1 = (lane < 16u) ? Mt[mr*3 + 1] : 0.0f;          // K=1 col | K=3 col (zero)
    }
    v2f av; av.x = a0; av.y = a1;

    // --- B operands (4x16 f32: VGPR0 = rowK0|rowK2, VGPR1 = rowK1|rowK3) ---
    unsigned j  = lane & 15u;
    unsigned n0 = nbase + j;        // group-0 pixel for column j
    unsigned n1 = n0 + 16u;        // group-1 pixel for column j
    v2f b0, b1;
    b0.x = (lane < 16u) ? (float)(n0 & (Ww - 1)) : 1.0f;  // x | homogeneous 1
    b0.y = (lane < 16u) ? (float)(n0 >> 8)       : 0.0f;  // y | zero pad
    b1.x = (lane < 16u) ? (float)(n1 & (Ww - 1)) : 1.0f;
    b1.y = (lane < 16u) ? (float)(n1 >> 8)       : 0.0f;

    v8f cz = {};
    v8f d0 = __builtin_amdgcn_wmma_f32_16x16x4_f32(false, av, false, b0,
                                                   (short)0, cz, false, false);
    v8f d1 = __builtin_amdgcn_wmma_f32_16x16x4_f32(false, av, false, b1,
                                                   (short)0, cz, false, false);

    float qx = (lane < 16u) ? d0[0] : d1[0];
    float qy = (lane < 16u) ? d0[1] : d1[1];
    float qz = (lane < 16u) ? d0[2] : d1[2];

    // --- finish projection per lane ---
    const float depth = depths[bv * Nn + n];
    const float prx = depth * qx + T0;
    const float pry = depth * qy + T1;
    const float z   = depth * qz + T2;
    const float zc  = (fabsf(z) > EPSV) ? z : EPSV;
    const float px  = prx / zc;
    const float py  = pry / zc;

    const bool valid = (z > EPSV) && (px > -1.0f) && (px < (float)Ww)
                                 && (py > -1.0f) && (py < (float)Hh);
    const float x0 = floorf(px), y0 = floorf(py);
    const float dx = px - x0,    dy = py - y0;

    const float cxs[4] = {x0, x0 + 1.0f, x0,        x0 + 1.0f};
    const float cys[4] = {y0, y0,        y0 + 1.0f, y0 + 1.0f};
    const float wts[4] = {(1.0f-dx)*(1.0f-dy), dx*(1.0f-dy), (1.0f-dx)*dy, dx*dy};
    int   idx4[4];
    float w4[4];
    #pragma unroll
    for (int k = 0; k < 4; ++k) {
        bool inb = valid && (cxs[k] >= 0.0f) && (cxs[k] <= (float)(Ww-1))
                         && (cys[k] >= 0.0f) && (cys[k] <= (float)(Hh-1));
        int xi = inb ? (int)cxs[k] : 0;
        int yi = inb ? (int)cys[k] : 0;
        idx4[k] = yi * Ww + xi;
        w4[k]   = inb ? wts[k] : 0.0f;
    }

    // --- scatter 64 feature channels (accumulators live in L2: 104MB < 192MB) ---
    const float* fin  = feats + (size_t)bv * Cc * Nn + n;
    float*       fout = out_f + (size_t)bv * Cc * Nn;
    #pragma unroll 4
    for (int c = 0; c < Cc; ++c) {
        float fv = fin[(size_t)c * Nn];
        if (c + 1 < Cc) __builtin_prefetch(fin + (size_t)(c + 1) * Nn, 0, 1);
        #pragma unroll
        for (int k = 0; k < 4; ++k)
            if (w4[k] != 0.0f)
                atomicAdd(fout + (size_t)c * Nn + idx4[k], fv * w4[k]);
    }
    // depth + weight channels
    float* dacc = out_d + (size_t)bv * Nn;
    float* wacc = wsum  + (size_t)bv * Nn;
    #pragma unroll
    for (int k = 0; k < 4; ++k) {
        if (w4[k] != 0.0f) {
            atomicAdd(dacc + idx4[k], z * w4[k]);
            atomicAdd(wacc + idx4[k], w4[k]);
        }
    }
}

// ---------------------------------------------------------------------------
// Kernel N: in-place normalize by (wsum + EPS)
// ---------------------------------------------------------------------------
__global__ void norm_kernel(float* __restrict__ out, const float* __restrict__ wsum) {
    const int FTOT = Bb * Vv * Cc * Nn;
    const int DTOT = Bb * Vv * Nn;
    int i = blockIdx.x * blockDim.x + threadIdx.x;
    if (i < FTOT) {
        int bv = i / (Cc * Nn);
        int n  = i % Nn;
        out[i] = out[i] / (wsum[bv * Nn + n] + EPSV);
    } else if (i < FTOT + DTOT) {
        int j = i - FTOT;
        out[i] = out[i] / (wsum[j] + EPSV);
    }
}

// ---------------------------------------------------------------------------
extern "C" void kernel_launch(void* const* d_in, const int* in_sizes, int n_in,
                              void* d_out, int out_size, void* d_ws, size_t ws_size,
                              hipStream_t stream) {
    const float* feats   = (const float*)d_in[0];
    const float* depths  = (const float*)d_in[1];
    const float* Kmat    = (const float*)d_in[2];
    const float* srcinv  = (const float*)d_in[4];
    const float* dstRT   = (const float*)d_in[5];

    float* out_f = (float*)d_out;                               // B*V*C*N
    float* out_d = out_f + (size_t)Bb * Vv * Cc * Nn;           // B*V*N
    float* wsum  = (float*)d_ws;                                // B*V*N
    float* mats  = wsum + (size_t)Bb * Vv * Nn;                 // B*V*12

    const int OUT_N = Bb*Vv*Cc*Nn + Bb*Vv*Nn;
    const int WS_N  = Bb*Vv*Nn;

    // 1) zero accumulators (d_out is poisoned by harness; must re-zero each call)
    {
        int n = OUT_N;  // >= WS_N
        zero_kernel<<<(n + 255) / 256, 256, 0, stream>>>(out_f, OUT_N, wsum, WS_N);
    }
    // 2) fused per-(b,v) transform matrices
    mats_kernel<<<1, 32, 0, stream>>>(Kmat, srcinv, dstRT, mats);
    // 3) WMMA transform + atomic splat: one thread per pixel, exact cover
    {
        int npix = Bb * Vv * Nn;          // 393216 = 1536 * 256
        splat_kernel<<<npix / 256, 256, 0, stream>>>(feats, depths, mats,
                                                     out_f, out_d, wsum);
    }
    // 4) in-place normalize
    norm_kernel<<<(OUT_N + 255) / 256, 256, 0, stream>>>(out_f, wsum);

    (void)in_sizes; (void)n_in; (void)out_size; (void)ws_size;
}